// SSIM3D_69234872812164
// MI455X (gfx1250) — compile-verified
//
#include <hip/hip_runtime.h>
#include <math.h>

// ---------------- CDNA5 (gfx1250) fused 3D-SSIM, v3 ----------------
// Raw img tiles in LDS once; stats on the fly feed banded-matmul blurs on
// V_WMMA_F32_16X16X4_F32 with dual accumulator chains; buf2 relayout so the
// D-pass A-fragments are adjacent-dword LDS pairs.

#define WS   11
#define RAD  5
#define TD   16              // output tile edge
#define REG  26              // TD + 2*RAD
#define NLINE1 (REG*REG)     // 676 lines for W-pass
#define NGRP1 43             // ceil(676/16)
#define NGRP2 26             // 416/16
#define NGRP3 16             // 256/16
#define PADL (NGRP1*16)      // 688: bufB padded so pass-1 stores are unguarded
#define S2   27              // buf2 line stride (pad: gcd(27,64)=1 -> conflict-free loads)
#define NBLOCKS (48*8*8)     // 3072

typedef float v2f __attribute__((ext_vector_type(2)));
typedef float v8f __attribute__((ext_vector_type(8)));

__device__ __forceinline__ int imin(int a, int b) { return a < b ? a : b; }

__global__ __launch_bounds__(256) void ssim3d_fused(
    const float* __restrict__ img1, const float* __restrict__ img2,
    float* __restrict__ partials)
{
    __shared__ float T1[REG * REG * REG];    // raw img1 halo region (70.3 KB)
    __shared__ float T2[REG * REG * REG];    // raw img2 halo region (70.3 KB)
    __shared__ float bufB[PADL * TD];        // pass-1 (W-blur) out, padded (44.0 KB)
    __shared__ float buf2[TD * TD * S2];     // pass-2 out, [h*16+w][d], stride 27 (27.6 KB)
    __shared__ float red[8];

    const int tid  = threadIdx.x;
    const int lane = tid & 31;
    const int wv   = tid >> 5;
    const int hi16 = (lane >> 4) & 1;        // 0: lanes 0-15, 1: lanes 16-31
    const int lm   = lane & 15;

    const int wT = blockIdx.x;               // 0..7
    const int hT = blockIdx.y;               // 0..7
    const int z  = blockIdx.z;               // 0..47 -> (b, c, dT)
    const int dT = z % 6;
    const int c  = (z / 6) % 4;
    const int b  = z / 24;

    const int gw0 = wT * TD - RAD;
    const int gh0 = hT * TD - RAD;
    const int gd0 = dT * TD - RAD;
    const long long base = (long long)(b * 4 + c) * 96LL * 128LL * 128LL;

    // Gaussian weights in fp64 exactly as the reference generates them:
    // mean = 11/2 = 5.5 (asymmetric), var = (11^2-1)/12 = 10, normalized.
    float g[WS];
    {
        double e[WS]; double ssum = 0.0;
#pragma unroll
        for (int t = 0; t < WS; ++t) {
            double x = (double)t - (double)WS * 0.5;
            e[t] = exp(-(x * x) / (2.0 * ((double)(WS * WS - 1) / 12.0)));
            ssum += e[t];
        }
#pragma unroll
        for (int t = 0; t < WS; ++t) g[t] = (float)(e[t] / ssum);
    }

    // Banded-Gaussian B matrix, K=28 in 7 chunks of 4.
    // 32-bit B 4x16 layout: lanes 0-15 hold rows K={kc*4,kc*4+1} at N=lane,
    // lanes 16-31 hold rows K={kc*4+2,kc*4+3} at N=lane-16.
    v2f Bm[7];
#pragma unroll
    for (int kc = 0; kc < 7; ++kc) {
        int k0 = kc * 4 + 2 * hi16;
        int d0 = k0 - lm, d1 = k0 + 1 - lm;
        Bm[kc].x = (d0 >= 0 && d0 <= 10) ? g[d0] : 0.0f;
        Bm[kc].y = (d1 >= 0 && d1 <= 10) ? g[d1] : 0.0f;
    }

    // ---- load both raw tiles ONCE (zero "SAME" padding outside the image) ----
    for (int t = tid; t < REG * REG * REG; t += 256) {
        int w = t % REG;
        int h = (t / REG) % REG;
        int d = t / (REG * REG);
        int gw = gw0 + w, gh = gh0 + h, gd = gd0 + d;
        float v1 = 0.f, v2 = 0.f;
        if ((unsigned)gw < 128u && (unsigned)gh < 128u && (unsigned)gd < 96u) {
            long long idx = base + ((long long)gd * 128 + gh) * 128 + gw;
            v1 = img1[idx];
            v2 = img2[idx];
        }
        T1[t] = v1;
        T2[t] = v2;
    }
    __syncthreads();

    v8f res[2][5];   // D-blurred tile per wave: 2 groups x 5 channels (80 VGPRs)

#pragma unroll
    for (int s = 0; s < 5; ++s) {
        // ---- pass 1: blur along W; stat generated on the fly from T1/T2 ----
        for (int grp = wv; grp < NGRP1; grp += 8) {
            v8f accE = 0.0f, accO = 0.0f;                 // dual chains for ILP
            int line = imin(grp * 16 + lm, NLINE1 - 1);   // clamp; results discarded
            int lb   = line * REG;
#pragma unroll
            for (int kc = 0; kc < 7; ++kc) {
                int k0 = kc * 4 + 2 * hi16;
                int o0 = lb + imin(k0,     REG - 1);      // pad K rows have B==0
                int o1 = lb + imin(k0 + 1, REG - 1);
                v2f a;
                if (s == 0)      { a.x = T1[o0];          a.y = T1[o1]; }
                else if (s == 1) { a.x = T2[o0];          a.y = T2[o1]; }
                else if (s == 2) { float p0 = T1[o0], p1 = T1[o1]; a.x = p0*p0; a.y = p1*p1; }
                else if (s == 3) { float q0 = T2[o0], q1 = T2[o1]; a.x = q0*q0; a.y = q1*q1; }
                else             { a.x = T1[o0]*T2[o0];   a.y = T1[o1]*T2[o1]; }
                if (kc & 1)
                    accO = __builtin_amdgcn_wmma_f32_16x16x4_f32(
                        false, a, false, Bm[kc], (short)0, accO, false, false);
                else
                    accE = __builtin_amdgcn_wmma_f32_16x16x4_f32(
                        false, a, false, Bm[kc], (short)0, accE, false, false);
            }
            v8f acc = accE + accO;
#pragma unroll
            for (int r = 0; r < 8; ++r)                   // padded bufB: no guards
                bufB[(grp * 16 + r + 8 * hi16) * TD + lm] = acc[r];
        }
        __syncthreads();   // bufB ready; also: buf2 readers (prev pass 3) are done

        // ---- pass 2: blur along H. lines=(d,w) 416 -> 26 groups ----
        for (int grp = wv; grp < NGRP2; grp += 8) {
            v8f accE = 0.0f, accO = 0.0f;
            int m  = grp * 16 + lm;                       // always < 416
            int dd = m >> 4, w = m & 15;
#pragma unroll
            for (int kc = 0; kc < 7; ++kc) {
                int k0 = kc * 4 + 2 * hi16;
                v2f a;
                a.x = bufB[(dd * REG + imin(k0,     REG - 1)) * TD + w];
                a.y = bufB[(dd * REG + imin(k0 + 1, REG - 1)) * TD + w];
                if (kc & 1)
                    accO = __builtin_amdgcn_wmma_f32_16x16x4_f32(
                        false, a, false, Bm[kc], (short)0, accO, false, false);
                else
                    accE = __builtin_amdgcn_wmma_f32_16x16x4_f32(
                        false, a, false, Bm[kc], (short)0, accE, false, false);
            }
            v8f acc = accE + accO;
#pragma unroll
            for (int r = 0; r < 8; ++r) {
                int mo = grp * 16 + r + 8 * hi16;
                int d2 = mo >> 4, w2 = mo & 15;
                buf2[(lm * TD + w2) * S2 + d2] = acc[r];  // [h=lm][w][d], stride 27
            }
        }
        __syncthreads();   // buf2 ready for pass 3

        // ---- pass 3: blur along D. lines=(h,w) 256 -> 16 groups; results in VGPRs ----
        int gi = 0;
        for (int grp = wv; grp < NGRP3; grp += 8, ++gi) { // exactly 2 iters per wave
            v8f accE = 0.0f, accO = 0.0f;
            int mb = (grp * 16 + lm) * S2;                // line3 = h*16+w = m directly
#pragma unroll
            for (int kc = 0; kc < 7; ++kc) {
                int k0 = kc * 4 + 2 * hi16;
                v2f a;
                a.x = buf2[mb + imin(k0,     REG - 1)];   // adjacent dwords -> paired load
                a.y = buf2[mb + imin(k0 + 1, REG - 1)];
                if (kc & 1)
                    accO = __builtin_amdgcn_wmma_f32_16x16x4_f32(
                        false, a, false, Bm[kc], (short)0, accO, false, false);
                else
                    accE = __builtin_amdgcn_wmma_f32_16x16x4_f32(
                        false, a, false, Bm[kc], (short)0, accE, false, false);
            }
            res[gi][s] = accE + accO;
        }
        // no trailing sync needed: next channel's pass 1 touches only bufB,
        // whose readers (this channel's pass 2) finished before the last sync.
    }

    // ---- SSIM combine + block reduction ----
    const float C1 = 1e-4f;   // 0.01^2
    const float C2 = 9e-4f;   // 0.03^2
    float part = 0.f;
#pragma unroll
    for (int gi = 0; gi < 2; ++gi) {
#pragma unroll
        for (int r = 0; r < 8; ++r) {
            float m1 = res[gi][0][r], m2 = res[gi][1][r];
            float x2 = res[gi][2][r], y2 = res[gi][3][r], xy = res[gi][4][r];
            float m1s = m1 * m1, m2s = m2 * m2, m12 = m1 * m2;
            float num = (2.f * m12 + C1) * (2.f * (xy - m12) + C2);
            float den = (m1s + m2s + C1) * ((x2 - m1s) + (y2 - m2s) + C2);
            part += num / den;
        }
    }
#pragma unroll
    for (int off = 16; off > 0; off >>= 1)
        part += __shfl_xor(part, off, 32);   // wave32 butterfly reduce
    if (lane == 0) red[wv] = part;
    __syncthreads();
    if (tid == 0) {
        float t = 0.f;
#pragma unroll
        for (int i = 0; i < 8; ++i) t += red[i];
        partials[(blockIdx.z * 8 + blockIdx.y) * 8 + blockIdx.x] = t;
    }
}

// Deterministic fixed-order final reduction (no float atomics).
__global__ __launch_bounds__(256) void ssim3d_reduce(
    const float* __restrict__ partials, float* __restrict__ out)
{
    __shared__ double redd[8];
    double acc = 0.0;
    for (int i = threadIdx.x; i < NBLOCKS; i += 256) acc += (double)partials[i];
#pragma unroll
    for (int off = 16; off > 0; off >>= 1)
        acc += __shfl_xor(acc, off, 32);
    int lane = threadIdx.x & 31, wv = threadIdx.x >> 5;
    if (lane == 0) redd[wv] = acc;
    __syncthreads();
    if (threadIdx.x == 0) {
        double t = 0.0;
#pragma unroll
        for (int i = 0; i < 8; ++i) t += redd[i];
        out[0] = (float)(t / 12582912.0);   // mean over 2*4*96*128*128
    }
}

extern "C" void kernel_launch(void* const* d_in, const int* in_sizes, int n_in,
                              void* d_out, int out_size, void* d_ws, size_t ws_size,
                              hipStream_t stream)
{
    (void)in_sizes; (void)n_in; (void)out_size; (void)ws_size;
    const float* img1 = (const float*)d_in[0];
    const float* img2 = (const float*)d_in[1];
    float* out   = (float*)d_out;
    float* parts = (float*)d_ws;             // 3072 floats of scratch

    dim3 grid(8, 8, 48);                     // (wTile, hTile, b*c*dTile)
    dim3 blk(256);                           // 8 wave32 per block
    ssim3d_fused<<<grid, blk, 0, stream>>>(img1, img2, parts);
    ssim3d_reduce<<<1, 256, 0, stream>>>(parts, out);
}